// BaseGINE_37374805410178
// MI455X (gfx1250) — compile-verified
//
#include <hip/hip_runtime.h>
#include <hip/hip_bf16.h>
#include <math.h>

typedef __attribute__((ext_vector_type(2))) float v2f;
typedef __attribute__((ext_vector_type(8))) float v8f;

#define NDIM 128
#define MTILE 80            // 50000 = 80 * 625 exactly; 5 subtiles of 16 per wave
#define GELU_C 0.70710678118654752f
#define INV_SQRT2 0.70710678118654752f

__device__ __forceinline__ float gelu_exact(float v) {
    return 0.5f * v * (1.0f + erff(v * GELU_C));
}

// ---------------------------------------------------------------- zero scratch
__global__ void zero_kernel(float* __restrict__ p, long long n) {
    long long i = (long long)blockIdx.x * blockDim.x + threadIdx.x;
    if (i < n) p[i] = 0.0f;
}

// ------------------------------------------------- edge message + scatter-add
// 2 edges per 256-thread block; thread d = feature. edge_emb recomputed from
// edge_attr @ We (We staged in LDS) to save bandwidth vs materializing it.
// agg (25.6 MB) is L2-resident, so the f32 atomics resolve in L2.
__global__ __launch_bounds__(256) void edge_message_kernel(
    const float* __restrict__ x, const long long* __restrict__ eidx,
    const float* __restrict__ eattr, const float* __restrict__ ew,
    const float* __restrict__ We, const float* __restrict__ be,
    float* __restrict__ agg, int E)
{
    __shared__ float sWe[16][NDIM];
    __shared__ float sEa[2][16];
    int tid = threadIdx.x;
    for (int i = tid; i < 16 * NDIM; i += 256)
        sWe[i >> 7][i & 127] = We[i];
    int e0 = blockIdx.x * 2;
    if (tid < 32) {
        int ee = e0 + (tid >> 4);
        if (ee < E) sEa[tid >> 4][tid & 15] = eattr[(size_t)ee * 16 + (tid & 15)];
    }
    __syncthreads();
    int le = tid >> 7;          // which of the 2 edges
    int d  = tid & 127;         // feature index
    int e  = e0 + le;
    if (e >= E) return;
    long long s = eidx[e];
    long long t = eidx[(size_t)E + e];
    float emb = be[d];
#pragma unroll
    for (int k = 0; k < 16; ++k) emb += sEa[le][k] * sWe[k][d];
    float m = x[(size_t)s * NDIM + d] + emb;
    m = gelu_exact(m) * ew[e];
    atomicAdd(&agg[(size_t)t * NDIM + d], m);
}

// ------------------------------------------------------- fused WMMA f32 GEMM
// out[M x 128] = act( A' @ W + bias ),  A' = (1+eps)*A + Aagg (optional fuse)
// 80-row M-tile per block; 8 waves = 8 N-tiles; each wave runs 5 independent
// accumulator chains (5 M-subtiles) on V_WMMA_F32_16X16X4_F32, with the
// B-fragment loaded once per k-step and reused across all 5 WMMAs.
__global__ __launch_bounds__(256) void gemm_fused_kernel(
    const float* __restrict__ A, const float* __restrict__ Aagg,
    const float* __restrict__ epsArr, int layer,
    const float* __restrict__ W, const float* __restrict__ bias,
    float* __restrict__ out, int applyGelu)
{
    __shared__ float sA[MTILE][132];  // stride 132: conflict-free A-frag reads
    __shared__ float sW[32][136];     // stride 136: half-wave K-offsets disjoint

    int tid  = threadIdx.x;
    int wave = tid >> 5;
    int lane = tid & 31;
    int row0 = blockIdx.x * MTILE;

    float epsv = 1.0f;
    if (epsArr) epsv = 1.0f + epsArr[layer];

    // Stage A tile (80 x 128), with (1+eps)*x + agg fusion
    for (int i = tid; i < MTILE * NDIM; i += 256) {
        int r = i >> 7, c = i & 127;
        float v = A[(size_t)(row0 + r) * NDIM + c];
        if (Aagg) v = epsv * v + Aagg[(size_t)(row0 + r) * NDIM + c];
        sA[r][c] = v;
    }

    v8f acc[5] = {};
    int n0   = wave * 16;
    int mrow = lane & 15;
    int koff = (lane >> 4) << 1;   // lanes 16-31 hold K+2 / K+3

    for (int kc = 0; kc < NDIM; kc += 32) {
        __syncthreads();
        for (int i = tid; i < 32 * NDIM; i += 256) {
            int r = i >> 7, c = i & 127;
            sW[r][c] = W[(size_t)(kc + r) * NDIM + c];
        }
        __syncthreads();
#pragma unroll
        for (int k = 0; k < 32; k += 4) {
            v2f b;
            b.x = sW[k + koff][n0 + mrow];
            b.y = sW[k + koff + 1][n0 + mrow];
#pragma unroll
            for (int t = 0; t < 5; ++t) {
                v2f a;
                a.x = sA[t * 16 + mrow][kc + k + koff];
                a.y = sA[t * 16 + mrow][kc + k + koff + 1];
                acc[t] = __builtin_amdgcn_wmma_f32_16x16x4_f32(
                    false, a, false, b, (short)0, acc[t], false, false);
            }
        }
    }

    // Epilogue: bias (+ optional exact GELU), write the 5 D tiles
    int col  = n0 + mrow;
    float bv = bias[col];
#pragma unroll
    for (int t = 0; t < 5; ++t) {
        int rbase = row0 + t * 16 + ((lane >> 4) << 3);  // lanes>=16 -> M+8
#pragma unroll
        for (int v = 0; v < 8; ++v) {
            float h = acc[t][v] + bv;
            if (applyGelu) h = gelu_exact(h);
            out[(size_t)(rbase + v) * NDIM + col] = h;
        }
    }
}

// ------------------------------------------------------------ BN column stats
__global__ __launch_bounds__(128) void bn_stats_kernel(
    const float* __restrict__ h, float* __restrict__ sums,
    float* __restrict__ sqs, int M)
{
    int c = threadIdx.x;
    float s = 0.0f, q = 0.0f;
    for (int r = blockIdx.x; r < M; r += gridDim.x) {
        float v = h[(size_t)r * NDIM + c];
        s += v; q += v * v;
    }
    atomicAdd(&sums[c], s);
    atomicAdd(&sqs[c], q);
}

// ------------------------------------- BN normalize + GELU + residual output
__global__ __launch_bounds__(256) void bn_act_residual_kernel(
    const float* __restrict__ xin, const float* __restrict__ h2,
    const float* __restrict__ sums, const float* __restrict__ sqs,
    const float* __restrict__ gamma, const float* __restrict__ beta,
    float* __restrict__ xout, int M)
{
    long long i = (long long)blockIdx.x * blockDim.x + threadIdx.x;
    long long total = (long long)M * NDIM;
    if (i >= total) return;
    int c = (int)(i & 127);
    float invM = 1.0f / (float)M;
    float mu  = sums[c] * invM;
    float var = sqs[c] * invM - mu * mu;
    float inv = rsqrtf(var + 1e-5f);
    float hv  = (h2[i] - mu) * inv * gamma[c] + beta[c];
    hv = gelu_exact(hv);
    xout[i] = (xin[i] + hv) * INV_SQRT2;
}

extern "C" void kernel_launch(void* const* d_in, const int* in_sizes, int n_in,
                              void* d_out, int out_size, void* d_ws, size_t ws_size,
                              hipStream_t stream) {
    const float*     x0    = (const float*)d_in[0];
    const long long* eidx  = (const long long*)d_in[1];
    const float*     eattr = (const float*)d_in[2];
    const float*     ew    = (const float*)d_in[3];
    const float*     We    = (const float*)d_in[4];
    const float*     be    = (const float*)d_in[5];
    const float*     W1    = (const float*)d_in[6];
    const float*     b1    = (const float*)d_in[7];
    const float*     W2    = (const float*)d_in[8];
    const float*     b2    = (const float*)d_in[9];
    const float*     eps   = (const float*)d_in[10];
    const float*     gamma = (const float*)d_in[11];
    const float*     beta  = (const float*)d_in[12];

    const int N = in_sizes[0] / NDIM;   // 50000
    const int E = in_sizes[3];          // 800000
    const long long ND = (long long)N * NDIM;

    // Workspace layout (floats): agg | sums | sumsq | h1 | h2 | xA | xB
    float* ws    = (float*)d_ws;
    float* agg   = ws;
    float* sums  = agg + ND;
    float* sqs   = sums + NDIM;
    float* h1    = sqs + NDIM;
    float* h2    = h1 + ND;
    float* xA    = h2 + ND;
    float* xB    = xA + ND;

    const long long zeroN = ND + 2 * NDIM;          // agg + sums + sumsq
    const int zeroBlocks  = (int)((zeroN + 255) / 256);
    const int edgeBlocks  = (E + 1) / 2;
    const int gemmBlocks  = N / MTILE;              // 625 exact
    const int elemBlocks  = (int)((ND + 255) / 256);

    const float* xcur = x0;
    for (int i = 0; i < 3; ++i) {
        float* xnext = (i == 0) ? xA : (i == 1) ? xB : (float*)d_out;

        zero_kernel<<<zeroBlocks, 256, 0, stream>>>(agg, zeroN);

        edge_message_kernel<<<edgeBlocks, 256, 0, stream>>>(
            xcur, eidx, eattr, ew, We, be, agg, E);

        gemm_fused_kernel<<<gemmBlocks, 256, 0, stream>>>(
            xcur, agg, eps, i, W1 + (size_t)i * NDIM * NDIM, b1 + i * NDIM,
            h1, /*gelu=*/1);

        gemm_fused_kernel<<<gemmBlocks, 256, 0, stream>>>(
            h1, nullptr, nullptr, 0, W2 + (size_t)i * NDIM * NDIM, b2 + i * NDIM,
            h2, /*gelu=*/0);

        bn_stats_kernel<<<512, 128, 0, stream>>>(h2, sums, sqs, N);

        bn_act_residual_kernel<<<elemBlocks, 256, 0, stream>>>(
            xcur, h2, sums, sqs, gamma + i * NDIM, beta + i * NDIM, xnext, N);

        xcur = xnext;
    }
}